// LogSumLayer_31696858644652
// MI455X (gfx1250) — compile-verified
//
#include <hip/hip_runtime.h>
#include <math.h>

#if __has_builtin(__builtin_amdgcn_global_load_async_to_lds_b128)
#warning "PROBE_HAS: __builtin_amdgcn_global_load_async_to_lds_b128"
#endif
#if __has_builtin(__builtin_amdgcn_s_wait_asynccnt)
#warning "PROBE_HAS: __builtin_amdgcn_s_wait_asynccnt"
#endif

// ---------------------------------------------------------------------------
// Problem constants
// ---------------------------------------------------------------------------
#define NUM_SEGMENTS (1 << 21)   // 2,097,152
#define NUM_ENTRIES  (1 << 24)   // 16,777,216
#define EPT          8           // entries per thread (avg segment length = 8)
#define TPB          256
#define LSE_EPS      1e-15f

// LDS staging: per-thread "row" of EPT entries = 64B, padded to 80B so b128
// readback is 2-phase conflict-free (banks 20*t mod 64).
#define ROW_BYTES    80
#define REGION_BYTES (TPB * ROW_BYTES)        // 20480 B per stream
#define CHUNKS       4                        // 4 x b128 per thread per stream

typedef long long v2ll __attribute__((ext_vector_type(2)));

// ---------------------------------------------------------------------------
// CDNA5 async global->LDS copy (ASYNCcnt-tracked), per CDNA5 ISA:
//   each lane copies 16 bytes from global sbase + goff to its LDS byte address
// ---------------------------------------------------------------------------
__device__ __forceinline__ void async_stage_b128(unsigned lds_byte_addr,
                                                 unsigned goff_bytes,
                                                 const void* sbase) {
    asm volatile("global_load_async_to_lds_b128 %0, %1, %2"
                 :
                 : "v"(lds_byte_addr), "v"(goff_bytes), "s"(sbase)
                 : "memory");
}

__device__ __forceinline__ void wait_asynccnt0() {
    asm volatile("s_wait_asynccnt 0x0" ::: "memory");
}

// Packed per-segment state in d_ws: hi 32 = running max M (f32 bits),
// lo 32 = running sum S = sum(exp(g - M)) (f32 bits). Init {M=-inf, S=0}.
__device__ __forceinline__ void lse_flush(unsigned long long* __restrict__ addr,
                                          float m, float s) {
    unsigned long long old = *addr;   // hint read; CAS loop validates
    unsigned long long assumed;
    do {
        assumed = old;
        const float m0 = __uint_as_float((unsigned int)(assumed >> 32));
        const float s0 = __uint_as_float((unsigned int)(assumed & 0xFFFFFFFFull));
        const float M  = fmaxf(m0, m);
        const float S  = s0 * __expf(m0 - M) + s * __expf(m - M); // exp(-inf)=0
        const unsigned long long packed =
            ((unsigned long long)__float_as_uint(M) << 32) |
             (unsigned long long)__float_as_uint(S);
        old = atomicCAS(addr, assumed, packed);
    } while (old != assumed);
}

__global__ void __launch_bounds__(TPB)
k_init(unsigned long long* __restrict__ seg_state) {
    const unsigned i = blockIdx.x * blockDim.x + threadIdx.x;
    seg_state[i] = 0xFF80000000000000ull;   // {M=-inf, S=0.0f}
}

__global__ void __launch_bounds__(TPB)
k_scan(const float* __restrict__ x,
       const long long* __restrict__ ptrs,
       const long long* __restrict__ csr,
       unsigned long long* __restrict__ seg_state) {
    __shared__ __align__(16) unsigned char stage[2 * REGION_BYTES];

    const unsigned tid = threadIdx.x;
    const size_t   blk_entry_base = (size_t)blockIdx.x * (TPB * EPT);
    const char*    csr_base = (const char*)(csr  + blk_entry_base);
    const char*    ptr_base = (const char*)(ptrs + blk_entry_base);
    const unsigned lds_base = (unsigned)(uintptr_t)&stage[0]; // LDS byte addr

    // Stage this block's 16KB csr chunk + 16KB ptrs chunk into LDS with
    // ASYNCcnt-tracked async copies (16B per lane per instruction).
    #pragma unroll
    for (int j = 0; j < CHUNKS; ++j) {
        const unsigned i16  = (unsigned)j * TPB + tid;                // 16B chunk idx
        const unsigned goff = i16 * 16u;                              // global byte off
        const unsigned loff = (i16 >> 2) * ROW_BYTES + (i16 & 3u) * 16u;
        async_stage_b128(lds_base + loff,                goff, csr_base);
        async_stage_b128(lds_base + REGION_BYTES + loff, goff, ptr_base);
    }
    wait_asynccnt0();    // my wave's asyncs done
    __syncthreads();     // everyone's asyncs done -> LDS fully populated

    // Read back my padded row: 4 x ds_load_b128 per stream, conflict-free.
    long long seg[EPT];
    long long p[EPT];
    #pragma unroll
    for (int k = 0; k < CHUNKS; ++k) {
        const size_t roff = (size_t)tid * ROW_BYTES + (size_t)k * 16u;
        v2ll c = *(const v2ll*)(stage + roff);
        seg[2 * k]     = c.x;
        seg[2 * k + 1] = c.y;
        v2ll q = *(const v2ll*)(stage + REGION_BYTES + roff);
        p[2 * k]     = q.x;
        p[2 * k + 1] = q.y;
    }

    // Independent gathers (x is 16MB -> L2-resident) for maximum MLP.
    float g[EPT];
    #pragma unroll
    for (int i = 0; i < EPT; ++i) g[i] = x[p[i]];

    // Online log-sum-exp over runs of equal segment id (csr is sorted).
    long long cur = seg[0];
    __builtin_prefetch(seg_state + cur, 0, 0);   // global_prefetch_b8
    float m = -__builtin_inff();
    float s = 0.0f;
    #pragma unroll
    for (int i = 0; i < EPT; ++i) {
        if (seg[i] != cur) {
            lse_flush(seg_state + cur, m, s);
            cur = seg[i];
            __builtin_prefetch(seg_state + cur, 0, 0);
            m = -__builtin_inff();
            s = 0.0f;
        }
        const float v = g[i];
        if (v > m) {
            s = s * __expf(m - v) + 1.0f;   // rescale old sum; exp(-inf)=0 seeds s=1
            m = v;
        } else {
            s += __expf(v - m);
        }
    }
    lse_flush(seg_state + cur, m, s);
}

__global__ void __launch_bounds__(TPB)
k_final(const unsigned long long* __restrict__ seg_state,
        float* __restrict__ out) {
    const unsigned i = blockIdx.x * blockDim.x + threadIdx.x;
    const unsigned long long v = seg_state[i];
    const float M = __uint_as_float((unsigned int)(v >> 32));
    const float S = __uint_as_float((unsigned int)(v & 0xFFFFFFFFull));
    // Empty segment: S=0, M=-inf  ->  log(eps) + (-inf) = -inf, matching JAX.
    out[i] = __logf(S + LSE_EPS) + M;
}

extern "C" void kernel_launch(void* const* d_in, const int* in_sizes, int n_in,
                              void* d_out, int out_size, void* d_ws, size_t ws_size,
                              hipStream_t stream) {
    (void)in_sizes; (void)n_in; (void)out_size; (void)ws_size;

    const float*     x    = (const float*)d_in[0];
    const long long* ptrs = (const long long*)d_in[1];
    const long long* csr  = (const long long*)d_in[2];
    float*           out  = (float*)d_out;

    unsigned long long* seg_state = (unsigned long long*)d_ws;  // 16 MB scratch

    const dim3 blk(TPB);
    k_init <<<NUM_SEGMENTS / TPB,        blk, 0, stream>>>(seg_state);
    k_scan <<<NUM_ENTRIES / (TPB * EPT), blk, 0, stream>>>(x, ptrs, csr, seg_state);
    k_final<<<NUM_SEGMENTS / TPB,        blk, 0, stream>>>(seg_state, out);
}